// GINConv_13400297963801
// MI455X (gfx1250) — compile-verified
//
#include <hip/hip_runtime.h>
#include <hip/hip_bf16.h>

#define N_NODES 10000
#define N_EDGES 640000
#define DFEAT   128
#define BN_EPS  1e-5f

typedef float v2f __attribute__((ext_vector_type(2)));
typedef float v8f __attribute__((ext_vector_type(8)));

// ---------------------------------------------------------------------------
// 0) zero the accumulator + degree regions of the workspace (must re-zero
//    every call: harness does not re-initialize between graph replays)
// ---------------------------------------------------------------------------
__global__ void zero_kernel(float* __restrict__ p, int n) {
    int i = blockIdx.x * blockDim.x + threadIdx.x;
    if (i < n) p[i] = 0.0f;
}

// ---------------------------------------------------------------------------
// 1) edge scatter: one wave32 per edge, lane l handles floats [4l..4l+3].
//    m_e = node[src_e] + edge_e accumulated into agg[dst_e] with f32 atomics
//    (GLOBAL_ATOMIC_ADD_F32 no-return -> STOREcnt). Lane 0 bumps degree.
//    node table (5 MB) and agg (5 MB) are L2-resident; edge_feats streams.
// ---------------------------------------------------------------------------
__global__ __launch_bounds__(256) void edge_scatter_kernel(
    const float* __restrict__ node, const float* __restrict__ edge,
    const int* __restrict__ src, const int* __restrict__ dst,
    float* __restrict__ agg, float* __restrict__ deg) {
    int gid  = blockIdx.x * blockDim.x + threadIdx.x;
    int e    = gid >> 5;          // wave32: one wave per edge
    int lane = gid & 31;
    if (e >= N_EDGES) return;     // uniform per-wave branch
    int s = src[e];
    int d = dst[e];
    const float4* np = (const float4*)(node + (size_t)s * DFEAT);
    const float4* ep = (const float4*)(edge + (size_t)e * DFEAT);
    float4 nv = np[lane];
    float4 ev = ep[lane];
    float* out = agg + (size_t)d * DFEAT + lane * 4;
    atomicAdd(out + 0, nv.x + ev.x);
    atomicAdd(out + 1, nv.y + ev.y);
    atomicAdd(out + 2, nv.z + ev.z);
    atomicAdd(out + 3, nv.w + ev.w);
    if (lane == 0) atomicAdd(deg + d, 1.0f);
}

// ---------------------------------------------------------------------------
// 2) GIN combine: h0 = (1+eps)*node + agg / max(deg,1)   (in place over agg)
// ---------------------------------------------------------------------------
__global__ void combine_kernel(const float* __restrict__ node,
                               const float* __restrict__ eps,
                               const float* __restrict__ deg,
                               float* __restrict__ h0 /* == agg */) {
    int i = blockIdx.x * blockDim.x + threadIdx.x;
    if (i >= N_NODES * DFEAT) return;
    int nidx = i >> 7;  // D == 128
    float dg = fmaxf(deg[nidx], 1.0f);
    h0[i] = (1.0f + eps[0]) * node[i] + h0[i] / dg;
}

// ---------------------------------------------------------------------------
// 3) GEMM Y = A(MxK) * B(KxN) + bias with V_WMMA_F32_16X16X4_F32.
//    K and N are compile-time -> fully resolvable K loop, no branches inside.
//    Each wave owns a 16x32 C tile (two 16x16 accumulators) so the A fragment
//    feeds two WMMAs per K-step (2x WMMA density per A load).
//    Fragment layouts (CDNA5 ISA):
//      A 16x4:  lane l -> row (l&15), K pair = k + 2*(l>>4)  (VGPR0=K, VGPR1=K+1)
//      B 4x16:  lane l -> col (l&15), same K-pair split per half-wave
//      C/D 16x16: VGPR r -> row r + 8*(l>>4), col = l&15
// ---------------------------------------------------------------------------
template <int KDIM, int NDIM>
__global__ __launch_bounds__(256) void gemm_wmma_kernel(
    const float* __restrict__ A, const float* __restrict__ B,
    const float* __restrict__ bias, float* __restrict__ Y,
    int total_tiles) {
    int wave = blockIdx.x * (blockDim.x >> 5) + (threadIdx.x >> 5);
    if (wave >= total_tiles) return;   // uniform per-wave; EXEC all-ones inside
    int lane = threadIdx.x & 31;

    constexpr int TN32 = NDIM / 32;    // 16x32 tiles along N
    int tm = wave / TN32;
    int tn = wave - tm * TN32;

    int half = lane >> 4;              // 0: K = k,k+1   1: K = k+2,k+3
    int l15  = lane & 15;
    int rowA = tm * 16 + l15;
    int col0 = tn * 32 + l15;
    int col1 = col0 + 16;
    const float* arow = A + (size_t)rowA * KDIM + 2 * half;

    v8f acc0 = {};
    v8f acc1 = {};
#pragma unroll 4
    for (int k = 0; k < KDIM; k += 4) {
        v2f a = *(const v2f*)(arow + k);             // contiguous 8B load
        const float* brow = B + (size_t)(k + 2 * half) * NDIM;
        v2f b0, b1;
        b0.x = brow[col0];
        b0.y = brow[NDIM + col0];
        b1.x = brow[col1];
        b1.y = brow[NDIM + col1];
        // 8 args: (neg_a, A, neg_b, B, c_mod, C, reuse_a, reuse_b)
        acc0 = __builtin_amdgcn_wmma_f32_16x16x4_f32(
            false, a, false, b0, (short)0, acc0, false, false);
        acc1 = __builtin_amdgcn_wmma_f32_16x16x4_f32(
            false, a, false, b1, (short)0, acc1, false, false);
    }

    float bv0 = bias[col0];
    float bv1 = bias[col1];
    float* yrow = Y + (size_t)(tm * 16 + 8 * half) * NDIM;
#pragma unroll
    for (int r = 0; r < 8; ++r) {
        yrow[(size_t)r * NDIM + col0] = acc0[r] + bv0;
        yrow[(size_t)r * NDIM + col1] = acc1[r] + bv1;
    }
}

// ---------------------------------------------------------------------------
// 4) per-column batch stats -> folded scale/bias
//    scale = gamma * rsqrt(var+eps); bias = beta - mean*scale
// ---------------------------------------------------------------------------
__global__ __launch_bounds__(256) void colstats_kernel(
    const float* __restrict__ X, const float* __restrict__ gamma,
    const float* __restrict__ beta, float* __restrict__ scale,
    float* __restrict__ bias, int rows, int cols) {
    int c = blockIdx.x;
    int t = threadIdx.x;
    float s = 0.0f, ss = 0.0f;
    for (int r = t; r < rows; r += blockDim.x) {
        float v = X[(size_t)r * cols + c];
        s += v;
        ss += v * v;
    }
    __shared__ float sh_s[256];
    __shared__ float sh_q[256];
    sh_s[t] = s;
    sh_q[t] = ss;
    __syncthreads();
    for (int o = 128; o > 0; o >>= 1) {
        if (t < o) { sh_s[t] += sh_s[t + o]; sh_q[t] += sh_q[t + o]; }
        __syncthreads();
    }
    if (t == 0) {
        float inv = 1.0f / (float)rows;
        float mu  = sh_s[0] * inv;
        float var = sh_q[0] * inv - mu * mu;
        float sc  = gamma[c] * rsqrtf(var + BN_EPS);
        scale[c] = sc;
        bias[c]  = beta[c] - mu * sc;
    }
}

// ---------------------------------------------------------------------------
// 5) fused BN(apply) + ReLU
// ---------------------------------------------------------------------------
__global__ void bnrelu_kernel(const float* __restrict__ X,
                              const float* __restrict__ scale,
                              const float* __restrict__ bias,
                              float* __restrict__ Yout, int n, int colmask) {
    int i = blockIdx.x * blockDim.x + threadIdx.x;
    if (i >= n) return;
    int c = i & colmask;  // cols is a power of two (256 / 128)
    float v = X[i] * scale[c] + bias[c];
    Yout[i] = fmaxf(v, 0.0f);
}

// ---------------------------------------------------------------------------
// launch
// ---------------------------------------------------------------------------
extern "C" void kernel_launch(void* const* d_in, const int* in_sizes, int n_in,
                              void* d_out, int out_size, void* d_ws, size_t ws_size,
                              hipStream_t stream) {
    const float* node   = (const float*)d_in[0];
    const float* edge   = (const float*)d_in[1];
    const float* eps    = (const float*)d_in[2];
    const float* W1     = (const float*)d_in[3];
    const float* b1     = (const float*)d_in[4];
    const float* gamma1 = (const float*)d_in[5];
    const float* beta1  = (const float*)d_in[6];
    const float* W2     = (const float*)d_in[7];
    const float* b2     = (const float*)d_in[8];
    const float* gamma2 = (const float*)d_in[9];
    const float* beta2  = (const float*)d_in[10];
    const int*   src    = (const int*)d_in[11];
    const int*   dst    = (const int*)d_in[12];
    float* out = (float*)d_out;

    // workspace layout (floats)
    float* ws     = (float*)d_ws;
    float* agg    = ws;                      // 1,280,000   (also h0, in place)
    float* deg    = ws + 1280000;            //    10,000
    float* y1     = ws + 1290240;            // 2,560,000   (also x1, in place)
    float* y2     = ws + 3850240;            // 1,280,000
    float* scale1 = ws + 5130240;            //       256
    float* bias1  = ws + 5130496;            //       256
    float* scale2 = ws + 5130752;            //       128
    float* bias2  = ws + 5130880;            //       128

    const int TB = 256;

    // 0) zero agg + deg (contiguous range)
    {
        int n = N_NODES * DFEAT + N_NODES;   // 1,290,000
        zero_kernel<<<(n + TB - 1) / TB, TB, 0, stream>>>(agg, n);
    }
    // 1) scatter edges
    {
        long long threads = (long long)N_EDGES * 32;
        edge_scatter_kernel<<<(int)((threads + TB - 1) / TB), TB, 0, stream>>>(
            node, edge, src, dst, agg, deg);
    }
    // 2) GIN combine (agg -> h0 in place)
    {
        int n = N_NODES * DFEAT;
        combine_kernel<<<(n + TB - 1) / TB, TB, 0, stream>>>(node, eps, deg, agg);
    }
    // 3) GEMM1: y1 = h0(10000x128) @ W1(128x256) + b1   (16x32 tile / wave)
    {
        int tiles = (N_NODES / 16) * (256 / 32);          // 625*8 = 5000 waves
        int blocks = (tiles + 7) / 8;                     // 8 waves per block
        gemm_wmma_kernel<128, 256><<<blocks, TB, 0, stream>>>(agg, W1, b1, y1,
                                                              tiles);
    }
    // 4) BN1 stats + apply + ReLU (in place on y1)
    colstats_kernel<<<256, TB, 0, stream>>>(y1, gamma1, beta1, scale1, bias1,
                                            N_NODES, 256);
    {
        int n = N_NODES * 256;
        bnrelu_kernel<<<(n + TB - 1) / TB, TB, 0, stream>>>(y1, scale1, bias1,
                                                            y1, n, 255);
    }
    // 5) GEMM2: y2 = x1(10000x256) @ W2(256x128) + b2
    {
        int tiles = (N_NODES / 16) * (DFEAT / 32);        // 625*4 = 2500 waves
        int blocks = (tiles + 7) / 8;
        gemm_wmma_kernel<256, 128><<<blocks, TB, 0, stream>>>(y1, W2, b2, y2,
                                                              tiles);
    }
    // 6) BN2 stats + apply + ReLU -> final output
    colstats_kernel<<<DFEAT, TB, 0, stream>>>(y2, gamma2, beta2, scale2, bias2,
                                              N_NODES, DFEAT);
    {
        int n = N_NODES * DFEAT;
        bnrelu_kernel<<<(n + TB - 1) / TB, TB, 0, stream>>>(y2, scale2, bias2,
                                                            out, n, DFEAT - 1);
    }
}